// SSM_12378095747352
// MI455X (gfx1250) — compile-verified
//
#include <hip/hip_runtime.h>

// Shapes from the reference
#define BLROWS 4096   // B_SZ * L
#define DDIM   1024   // D
#define NDIM   16     // N

typedef __attribute__((ext_vector_type(2))) float v2f;
typedef __attribute__((ext_vector_type(4))) float v4f;
typedef __attribute__((ext_vector_type(8))) float v8f;

__device__ __forceinline__ float softplusf(float v) {
    // numerically stable log1p(exp(v))
    return (v > 0.f) ? v + log1pf(__expf(-v)) : log1pf(__expf(v));
}

// ---------------------------------------------------------------------------
// Kernel 1: projections B = x*Wb^T, C = x*Wc^T via V_WMMA_F32_16X16X4_F32,
//           s_delta = x*Wd^T accumulated in VALU alongside.
// One wave (32 lanes) per 16-row tile of x. 256 tiles total.
// f32 WMMA layout (ISA 7.12.2):
//   A (16x4):  lane<16 -> M=lane, K={k,k+1}; lane>=16 -> M=lane-16, K={k+2,k+3}
//   B (4x16):  lane<16 -> N=lane, K={k,k+1}; lane>=16 -> N=lane-16, K={k+2,k+3}
//   C/D 16x16: vgpr i -> M = i + (lane<16 ? 0 : 8), N = lane%16
// ---------------------------------------------------------------------------
__global__ __launch_bounds__(128) void ssm_proj_wmma(
    const float* __restrict__ x,   // [BLROWS, DDIM]
    const float* __restrict__ Wb,  // [NDIM, DDIM]
    const float* __restrict__ Wc,  // [NDIM, DDIM]
    const float* __restrict__ Wd,  // [1, DDIM]
    float* __restrict__ Bbuf,      // [BLROWS, NDIM]
    float* __restrict__ Cbuf,      // [BLROWS, NDIM]
    float* __restrict__ sDelta)    // [BLROWS]
{
    const int lane = threadIdx.x & 31;
    const int wave = threadIdx.x >> 5;
    const int tile = blockIdx.x * 4 + wave;      // 16-row tile index, 0..255
    const int m    = lane & 15;                  // row-in-tile (A) / column N (B)
    const int koff = (lane >> 4) << 1;           // 0 for lanes 0-15, 2 for 16-31

    const float* xrow  = x  + (size_t)(tile * 16 + m) * DDIM;
    const float* wbrow = Wb + (size_t)m * DDIM;
    const float* wcrow = Wc + (size_t)m * DDIM;

    v8f accB = {};
    v8f accC = {};
    float sacc = 0.f;

    for (int k = 0; k < DDIM; k += 4) {
        const int kb = k + koff;
        v2f a  = *(const v2f*)(xrow  + kb);
        v2f bb = *(const v2f*)(wbrow + kb);
        v2f bc = *(const v2f*)(wcrow + kb);
        v2f wd = *(const v2f*)(Wd    + kb);

        accB = __builtin_amdgcn_wmma_f32_16x16x4_f32(
                   false, a, false, bb, (short)0, accB, false, false);
        accC = __builtin_amdgcn_wmma_f32_16x16x4_f32(
                   false, a, false, bc, (short)0, accC, false, false);

        sacc += a.x * wd.x + a.y * wd.y;
    }

    // D-matrix writeback: vgpr i holds row M = i + (lane<16 ? 0 : 8)
    const int mbase = tile * 16 + ((lane < 16) ? 0 : 8);
#pragma unroll
    for (int i = 0; i < 8; ++i) {
        Bbuf[(size_t)(mbase + i) * NDIM + m] = accB[i];
        Cbuf[(size_t)(mbase + i) * NDIM + m] = accC[i];
    }

    // lanes (l, l^16) together cover all K of row l%16
    float other = __shfl(sacc, lane ^ 16, 32);
    if (lane < 16) sDelta[tile * 16 + lane] = sacc + other;
}

// ---------------------------------------------------------------------------
// Kernel 2: rA = 1/A (kills per-element divides in the hot loop) and copy the
//           C rows into the output slot c == 2*D (streamed, non-temporal).
// ---------------------------------------------------------------------------
__global__ __launch_bounds__(256) void ssm_prep(
    const float* __restrict__ A,     // [DDIM, NDIM]
    float* __restrict__ rA,          // [DDIM, NDIM]
    const float* __restrict__ Cbuf,  // [BLROWS, NDIM]
    float* __restrict__ out)         // [BLROWS, 2*DDIM+1, NDIM]
{
    const int i = blockIdx.x * blockDim.x + threadIdx.x;
    if (i < DDIM * NDIM) rA[i] = 1.0f / A[i];
    if (i < BLROWS * NDIM) {
        const int r = i >> 4, n = i & 15;
        __builtin_nontemporal_store(
            Cbuf[i], out + ((size_t)r * (2 * DDIM + 1) + 2 * DDIM) * NDIM + n);
    }
}

// ---------------------------------------------------------------------------
// Kernel 3: the memory-bound bulk. One thread per (row r, channel d):
//   Delta    = softplus(sDelta[r] + delta_param[d])
//   A_bar[n] = exp(Delta * A[d][n])
//   dBx[n]   = (A_bar[n] - 1) * B[r][n] * x[r][d] / A[d][n]
// Each thread stores 2 x 64B contiguous chunks; adjacent threads adjacent
// chunks -> fully coalesced against the 23.3 TB/s HBM. The 537 MB output is
// write-once / no-reuse, so stores use the NT temporal hint to stream past
// L2 and keep A/rA/Bbuf resident. x is read once here -> NT load.
// ---------------------------------------------------------------------------
__global__ __launch_bounds__(256) void ssm_main(
    const float* __restrict__ x,       // [BLROWS, DDIM]
    const float* __restrict__ A,       // [DDIM, NDIM]
    const float* __restrict__ rA,      // [DDIM, NDIM]
    const float* __restrict__ dparam,  // [DDIM]
    const float* __restrict__ Bbuf,    // [BLROWS, NDIM]
    const float* __restrict__ sDelta,  // [BLROWS]
    float* __restrict__ out)           // [BLROWS, 2*DDIM+1, NDIM]
{
    const int r = blockIdx.y;
    const int d = blockIdx.x * blockDim.x + threadIdx.x;

    const float Delta = softplusf(sDelta[r] + dparam[d]);
    const float xv    = __builtin_nontemporal_load(x + (size_t)r * DDIM + d);

    const v4f* Ar  = (const v4f*)(A    + (size_t)d * NDIM);
    const v4f* RAr = (const v4f*)(rA   + (size_t)d * NDIM);
    const v4f* Br  = (const v4f*)(Bbuf + (size_t)r * NDIM);

    v4f* outA = (v4f*)(out + ((size_t)r * (2 * DDIM + 1) + d)        * NDIM);
    v4f* outD = (v4f*)(out + ((size_t)r * (2 * DDIM + 1) + DDIM + d) * NDIM);

#pragma unroll
    for (int i = 0; i < 4; ++i) {
        const v4f a  = Ar[i];
        const v4f ra = RAr[i];
        const v4f b  = Br[i];
        v4f abv, dbv;
        abv.x = __expf(Delta * a.x);
        abv.y = __expf(Delta * a.y);
        abv.z = __expf(Delta * a.z);
        abv.w = __expf(Delta * a.w);
        dbv.x = (abv.x - 1.f) * (b.x * ra.x) * xv;
        dbv.y = (abv.y - 1.f) * (b.y * ra.y) * xv;
        dbv.z = (abv.z - 1.f) * (b.z * ra.z) * xv;
        dbv.w = (abv.w - 1.f) * (b.w * ra.w) * xv;
        __builtin_nontemporal_store(abv, outA + i);
        __builtin_nontemporal_store(dbv, outD + i);
    }
}

// ---------------------------------------------------------------------------
extern "C" void kernel_launch(void* const* d_in, const int* in_sizes, int n_in,
                              void* d_out, int out_size, void* d_ws, size_t ws_size,
                              hipStream_t stream) {
    const float* x  = (const float*)d_in[0];  // [2,2048,1024]
    const float* Wb = (const float*)d_in[1];  // [16,1024]
    const float* Wc = (const float*)d_in[2];  // [16,1024]
    const float* Wd = (const float*)d_in[3];  // [1,1024]
    const float* A  = (const float*)d_in[4];  // [1024,16]
    const float* dp = (const float*)d_in[5];  // [1024]
    float* out = (float*)d_out;

    float* ws     = (float*)d_ws;
    float* Bbuf   = ws;                        // 65536 floats
    float* Cbuf   = ws + 65536;                // 65536 floats
    float* sDelta = ws + 131072;               // 4096 floats
    float* rA     = ws + 135168;               // 16384 floats (592 KB total)

    // 256 row-tiles, 4 waves per block
    ssm_proj_wmma<<<64, 128, 0, stream>>>(x, Wb, Wc, Wd, Bbuf, Cbuf, sDelta);
    // covers max(DDIM*NDIM, BLROWS*NDIM) = 65536 threads
    ssm_prep<<<256, 256, 0, stream>>>(A, rA, Cbuf, out);
    // one thread per (r, d)
    ssm_main<<<dim3(DDIM / 256, BLROWS), 256, 0, stream>>>(x, A, rA, dp, Bbuf, sDelta, out);
}